// Encoder_23407571763908
// MI455X (gfx1250) — compile-verified
//
#include <hip/hip_runtime.h>

// ---------------------------------------------------------------------------
// GCN encoder: e1 = A*e0 ; e2 = A*e1 ; summed = e0+e1+e2
// A given as COO (row_idx, col_idx, vals), NNZ=3.2M, dense = [N=100001, D=64] f32.
// d_out = [summed | e0 | e1 | e2], each N*D floats.
//
// MI455X reasoning: 0.82 GFLOP total vs ~200 MB of traffic -> pure memory /
// atomics bound. Dense matrix (25.6 MB) fits in 192 MB L2, so gathers and
// scatter-adds are L2-resident. SpMM uses the minimal wave32 instruction
// shape (half-wave per edge: 1x b128 gather + 4x global_atomic_add_f32,
// forced via inline asm at device scope). The dense sum epilogue runs on the
// matrix pipe as chained rank-4 shifted-identity WMMAs (exact result).
// ---------------------------------------------------------------------------

typedef __attribute__((ext_vector_type(2))) float v2f;
typedef __attribute__((ext_vector_type(8))) float v8f;

#define D_DIM 64

// Guaranteed hardware f32 atomic add, device scope, no return (STOREcnt path).
__device__ __forceinline__ void atomic_add_f32(float* p, float v) {
    asm volatile("global_atomic_add_f32 %0, %1, off scope:SCOPE_DEV"
                 :
                 : "v"(p), "v"(v)
                 : "memory");
}

// ---------------------------------------------------------------------------
// init: e1 = 0 ; e2 = 0   (float4 vectorized; e1,e2 are adjacent -> one range)
// ---------------------------------------------------------------------------
__global__ void zero_kernel(float4* __restrict__ dst, int n4) {
    int i = blockIdx.x * blockDim.x + threadIdx.x;
    if (i < n4) dst[i] = make_float4(0.f, 0.f, 0.f, 0.f);
}

// ---------------------------------------------------------------------------
// COO SpMM scatter: half-wave (16 lanes) per edge.
// Each lane: one 128-bit gather of x[col] (16 lanes * float4 = 256B row),
// scale by vals[e], 4 x global_atomic_add_f32 into y[row].
// ---------------------------------------------------------------------------
__global__ void spmm_kernel(const int*   __restrict__ row_idx,
                            const int*   __restrict__ col_idx,
                            const float* __restrict__ vals,
                            const float* __restrict__ x,
                            float*       __restrict__ y,
                            int nnz) {
    long long tid = (long long)blockIdx.x * blockDim.x + threadIdx.x;
    int e = (int)(tid >> 4);   // edge handled by this half-wave
    int l = (int)(tid & 15);   // float4 slot within the 64-wide row
    if (e >= nnz) return;

    int   r = row_idx[e];
    int   c = col_idx[e];
    float v = vals[e];

    const float4* xr = (const float4*)x + ((size_t)c * (D_DIM / 4) + l);
    float4 xv = *xr;                       // global_load_b128

    float* yp = y + ((size_t)r * D_DIM + l * 4);
    atomic_add_f32(yp + 0, v * xv.x);
    atomic_add_f32(yp + 1, v * xv.y);
    atomic_add_f32(yp + 2, v * xv.z);
    atomic_add_f32(yp + 3, v * xv.w);
}

// ---------------------------------------------------------------------------
// WMMA sum epilogue: summed = e0 + e1 + e2 (and streams out the e0 copy).
// One 16x16 tile per wave. Adding a 16x16 tile Y = 4 chained rank-4 updates:
//   D = A_j x B_j + C,  A_j[m][k] = (m == 4j+k) ? 1 : 0,  B_j = rows 4j..4j+3
// C seeded with the e0 tile -> 8 x v_wmma_f32_16x16x4_f32 per tile.
// FULL=true: no clamping/predication (6250 of 6251 row tiles).
// FULL=false: tail tile; clamp loads to row N-1 (clamped rows only influence
// unstored D rows) and predicate stores after all WMMAs (EXEC uniform around
// every WMMA, as the ISA requires).
// ---------------------------------------------------------------------------
template <bool FULL>
__device__ __forceinline__ void sum_tile(const float* __restrict__ e0,
                                         const float* __restrict__ e1,
                                         const float* __restrict__ e2,
                                         float* __restrict__ summed,
                                         float* __restrict__ e0_out,
                                         int r0, int c0, int m, int g,
                                         int n_rows) {
    // C = e0 tile in C/D layout: acc[q] = e0[r0 + q + 8g][c0 + m]
    v8f acc;
#pragma unroll
    for (int q = 0; q < 8; ++q) {
        int rr = r0 + q + 8 * g;
        int rc = FULL ? rr : (rr < n_rows ? rr : n_rows - 1);
        float t = e0[(size_t)rc * D_DIM + c0 + m];
        acc[q] = t;
        if (FULL || rr < n_rows)
            e0_out[(size_t)rr * D_DIM + c0 + m] = t;   // fused e0 copy
    }

    const float* mats[2] = { e1, e2 };
#pragma unroll
    for (int s = 0; s < 2; ++s) {
        const float* __restrict__ Y = mats[s];
#pragma unroll
        for (int j = 0; j < 4; ++j) {
            v2f a, b;
#pragma unroll
            for (int p = 0; p < 2; ++p) {
                int k = 2 * g + p;                       // K index for this lane/reg
                a[p] = (m == 4 * j + k) ? 1.0f : 0.0f;   // A_j[m][k]
                int rr = r0 + 4 * j + k;                 // B_j[k][n] = Y[r0+4j+k][c0+n]
                int rc = FULL ? rr : (rr < n_rows ? rr : n_rows - 1);
                b[p] = Y[(size_t)rc * D_DIM + c0 + m];
            }
            acc = __builtin_amdgcn_wmma_f32_16x16x4_f32(false, a, false, b,
                                                        (short)0, acc, false, false);
        }
    }

#pragma unroll
    for (int q = 0; q < 8; ++q) {
        int rr = r0 + q + 8 * g;
        if (FULL || rr < n_rows)
            summed[(size_t)rr * D_DIM + c0 + m] = acc[q];
    }
}

__global__ void wmma_sum_kernel(const float* __restrict__ e0,
                                const float* __restrict__ e1,
                                const float* __restrict__ e2,
                                float*       __restrict__ summed,
                                float*       __restrict__ e0_out,
                                int n_rows, int n_tiles) {
    int wave = (blockIdx.x * blockDim.x + threadIdx.x) >> 5;   // wave32
    if (wave >= n_tiles) return;                               // wave-uniform

    int tm = wave >> 2;            // D_DIM/16 == 4 column tiles
    int tn = wave & 3;
    int r0 = tm * 16;
    int c0 = tn * 16;

    int lane = threadIdx.x & 31;
    int m = lane & 15;             // N index for B/C/D; M index for A
    int g = lane >> 4;             // half-wave selector

    if (r0 + 16 <= n_rows) {       // wave-uniform fast path
        sum_tile<true>(e0, e1, e2, summed, e0_out, r0, c0, m, g, n_rows);
    } else {
        sum_tile<false>(e0, e1, e2, summed, e0_out, r0, c0, m, g, n_rows);
    }
}

// ---------------------------------------------------------------------------
extern "C" void kernel_launch(void* const* d_in, const int* in_sizes, int n_in,
                              void* d_out, int out_size, void* d_ws, size_t ws_size,
                              hipStream_t stream) {
    (void)n_in; (void)d_ws; (void)ws_size; (void)out_size;

    const int*   row_idx = (const int*)d_in[0];
    const int*   col_idx = (const int*)d_in[1];
    const float* vals    = (const float*)d_in[2];
    const float* emb     = (const float*)d_in[3];

    const int nnz = in_sizes[0];
    const int N   = in_sizes[3] / D_DIM;          // 100001
    const size_t ND = (size_t)N * D_DIM;

    float* out    = (float*)d_out;
    float* summed = out;
    float* e0_out = out + ND;
    float* e1     = out + 2 * ND;
    float* e2     = out + 3 * ND;

    // 1) zero the two accumulator regions (e1|e2 are contiguous)
    {
        int n4 = (int)(2 * ND / 4);
        int blocks = (n4 + 255) / 256;
        zero_kernel<<<blocks, 256, 0, stream>>>((float4*)e1, n4);
    }

    // 2) e1 = A * e0   (half-wave per edge)
    {
        long long threads = (long long)nnz * 16;
        int blocks = (int)((threads + 255) / 256);
        spmm_kernel<<<blocks, 256, 0, stream>>>(row_idx, col_idx, vals, emb, e1, nnz);
    }

    // 3) e2 = A * e1
    {
        long long threads = (long long)nnz * 16;
        int blocks = (int)((threads + 255) / 256);
        spmm_kernel<<<blocks, 256, 0, stream>>>(row_idx, col_idx, vals, e1, e2, nnz);
    }

    // 4) summed = e0 + e1 + e2 via chained rank-4 WMMA accumulation
    //    (also streams the e0 copy into d_out)
    {
        int n_row_tiles = (N + 15) / 16;                  // 6251
        int n_tiles = n_row_tiles * (D_DIM / 16);         // 25004 (1 tile per wave)
        int waves_per_block = 256 / 32;
        int blocks = (n_tiles + waves_per_block - 1) / waves_per_block;
        wmma_sum_kernel<<<blocks, 256, 0, stream>>>(emb, e1, e2, summed, e0_out,
                                                    N, n_tiles);
    }
}